// VOXCPM_FEAT_DECODER_47296179864185
// MI455X (gfx1250) — compile-verified
//
#include <hip/hip_runtime.h>
#include <hip/hip_bf16.h>
#include <cstdint>

// ---------------------------------------------------------------------------
// VoxCPM feat-decoder for MI455X (gfx1250).
// Bandwidth-bound: ~477MB of f32 weights per step -> stream them once through
// V_WMMA_F32_16X16X4_F32 (f32 in, f32 acc; no conversion cost).
// M is padded 18 -> 32 rows; one wave owns a 32x16 output tile (2 WMMA accs).
// Activations staged into LDS via global_load_async_to_lds_b128 (ASYNCcnt),
// weights streamed from global with strength-reduced pointer bumps.
// ---------------------------------------------------------------------------

typedef __attribute__((ext_vector_type(2))) float v2f;
typedef __attribute__((ext_vector_type(4))) float v4f;
typedef __attribute__((ext_vector_type(8))) float v8f;

#define HDIM  1024
#define NHEAD 16
#define KVH   2
#define DHEAD 64
#define IDIM  4096
#define QLEN  9
#define ROWS  18        // 2 batches * 9 tokens
#define MPAD  32        // padded M for WMMA tiles
#define CHUNK 256       // K-chunk staged in LDS
#define LDST  260       // padded LDS row stride (floats) -> conflict-free

// Async-stage one 32 x CHUNK f32 tile of A into LDS. Generic->LDS pointer
// truncation is valid: the LDS offset is the low 32 bits of the flat address.
__device__ __forceinline__ void stage_async(const float* __restrict__ A,
                                            float* As, int K, int k0) {
  for (int it = threadIdx.x; it < MPAD * (CHUNK / 4); it += 256) {
    int r  = it / (CHUNK / 4);
    int c4 = (it % (CHUNK / 4)) * 4;
    unsigned lds_off = (unsigned)(uintptr_t)(&As[r * LDST + c4]);
    const float* gp = A + (size_t)r * K + k0 + c4;
    asm volatile("global_load_async_to_lds_b128 %0, %1, off"
                 :: "v"(lds_off), "v"(gp) : "memory");
  }
  asm volatile("s_wait_asynccnt 0x0" ::: "memory");
}

// ---------------------------------------------------------------------------
// Prologue: build h (32 x 1024, rows 18..31 zero) and zero attn-out pad rows.
// ---------------------------------------------------------------------------
__global__ void prep_kernel(const float* __restrict__ random,
                            const float* __restrict__ dit_hidden,
                            const float* __restrict__ feat_cond,
                            const float* __restrict__ t_all,
                            const float* __restrict__ W_in,
                            const float* __restrict__ b_in,
                            const int*   __restrict__ step,
                            float* __restrict__ h,
                            float* __restrict__ ao) {
  int c = blockIdx.x * blockDim.x + threadIdx.x;
  if (c >= HDIM) return;
  int s = step[0];
  float tc = t_all[s * HDIM + c];
  h[0 * HDIM + c] = dit_hidden[c] + tc;   // b=0 token 0
  h[9 * HDIM + c] = tc;                   // b=1 token 0
  for (int p = 0; p < 4; ++p) {
    h[(1 + p)  * HDIM + c] = feat_cond[p * HDIM + c];
    h[(10 + p) * HDIM + c] = feat_cond[(4 + p) * HDIM + c];
  }
  for (int p = 0; p < 4; ++p) {
    float acc = b_in[c];
    for (int f = 0; f < 64; ++f) acc += random[p * 64 + f] * W_in[f * HDIM + c];
    h[(5 + p)  * HDIM + c] = acc;         // x duplicated in both batches
    h[(14 + p) * HDIM + c] = acc;
  }
  for (int r = ROWS; r < MPAD; ++r) {
    h[r * HDIM + c]  = 0.f;
    ao[r * HDIM + c] = 0.f;
  }
}

// ---------------------------------------------------------------------------
// RMSNorm over H per row (32 blocks; pad rows stay zero).
// ---------------------------------------------------------------------------
__global__ void rms_kernel(const float* __restrict__ x,
                           const float* __restrict__ w,
                           float* __restrict__ y) {
  __shared__ float red[256];
  int r = blockIdx.x;
  const float* xr = x + (size_t)r * HDIM;
  float ss = 0.f;
  for (int c = threadIdx.x; c < HDIM; c += 256) { float v = xr[c]; ss += v * v; }
  red[threadIdx.x] = ss;
  __syncthreads();
  for (int s2 = 128; s2 > 0; s2 >>= 1) {
    if ((int)threadIdx.x < s2) red[threadIdx.x] += red[threadIdx.x + s2];
    __syncthreads();
  }
  float inv = rsqrtf(red[0] * (1.f / HDIM) + 1e-6f);
  for (int c = threadIdx.x; c < HDIM; c += 256)
    y[(size_t)r * HDIM + c] = xr[c] * inv * w[c];
}

// Final RMS: rows 0..7 = tokens 5..8 of each batch; rows 8..31 zero.
__global__ void rms_final_kernel(const float* __restrict__ h,
                                 const float* __restrict__ w,
                                 float* __restrict__ y) {
  __shared__ float red[256];
  int r = blockIdx.x;
  if (r >= 8) {
    for (int c = threadIdx.x; c < HDIM; c += 256) y[(size_t)r * HDIM + c] = 0.f;
    return;
  }
  int m = (r >> 2) * QLEN + 5 + (r & 3);
  const float* xr = h + (size_t)m * HDIM;
  float ss = 0.f;
  for (int c = threadIdx.x; c < HDIM; c += 256) { float v = xr[c]; ss += v * v; }
  red[threadIdx.x] = ss;
  __syncthreads();
  for (int s2 = 128; s2 > 0; s2 >>= 1) {
    if ((int)threadIdx.x < s2) red[threadIdx.x] += red[threadIdx.x + s2];
    __syncthreads();
  }
  float inv = rsqrtf(red[0] * (1.f / HDIM) + 1e-6f);
  for (int c = threadIdx.x; c < HDIM; c += 256)
    y[(size_t)r * HDIM + c] = xr[c] * inv * w[c];
}

// ---------------------------------------------------------------------------
// Generic WMMA GEMM: C[32 x N] (+= / =) A[32 x K] * B[K x N] (+ bias).
// flags bit0: residual accumulate into C; bit1: add bias[n].
// One wave per 16-wide N tile; two 16x16 f32 accumulators cover M=32.
// ---------------------------------------------------------------------------
__global__ __launch_bounds__(256)
void gemm_kernel(const float* __restrict__ A, const float* __restrict__ B,
                 float* __restrict__ C, const float* __restrict__ bias,
                 int K, int N, int flags) {
  __shared__ float As[MPAD * LDST];
  const int lane = threadIdx.x & 31;
  const int wave = threadIdx.x >> 5;
  const int loN  = lane & 15;
  const int hi   = lane >> 4;             // 0|1 -> K sub-pair
  const int n0   = (blockIdx.x * 8 + wave) * 16;
  const bool act = (n0 < N);

  // Per-wave weight stream pointer: rows (k + 2*hi), column n0+loN.
  const float* Bp = B + (size_t)(2 * hi) * N + n0 + loN;
  const size_t bstep = (size_t)4 * N;

  v8f c0 = {}; v8f c1 = {};
  for (int k0 = 0; k0 < K; k0 += CHUNK) {
    stage_async(A, As, K, k0);
    __syncthreads();
    if (act) {
      const float* Ap = &As[loN * LDST];
#pragma unroll 4
      for (int kk = 0; kk < CHUNK; kk += 4) {
        int ka = kk + 2 * hi;
        v2f a0 = *(const v2f*)(Ap + ka);
        v2f a1 = *(const v2f*)(Ap + 16 * LDST + ka);
        v2f b;
        b.x = Bp[0];
        b.y = Bp[N];
        Bp += bstep;
        c0 = __builtin_amdgcn_wmma_f32_16x16x4_f32(false, a0, false, b, (short)0, c0, false, false);
        c1 = __builtin_amdgcn_wmma_f32_16x16x4_f32(false, a1, false, b, (short)0, c1, false, false);
      }
    }
    __syncthreads();
  }
  if (act) {
    float bv = (flags & 2) ? bias[n0 + loN] : 0.f;
#pragma unroll
    for (int r = 0; r < 8; ++r) {
      int m0 = r + 8 * hi;
      int m1 = m0 + 16;
      size_t i0 = (size_t)m0 * N + n0 + loN;
      size_t i1 = (size_t)m1 * N + n0 + loN;
      float v0 = c0[r] + bv, v1 = c1[r] + bv;
      if (flags & 1) { v0 += C[i0]; v1 += C[i1]; }
      C[i0] = v0; C[i1] = v1;
    }
  }
}

// ---------------------------------------------------------------------------
// Fused gate/up: T1 = silu(A*Wg) * (A*Wu). Same tiling, 4 WMMAs per K-step.
// ---------------------------------------------------------------------------
__global__ __launch_bounds__(256)
void gemm_gate_up_kernel(const float* __restrict__ A,
                         const float* __restrict__ Bg,
                         const float* __restrict__ Bu,
                         float* __restrict__ T1, int K, int N) {
  __shared__ float As[MPAD * LDST];
  const int lane = threadIdx.x & 31;
  const int wave = threadIdx.x >> 5;
  const int loN  = lane & 15;
  const int hi   = lane >> 4;
  const int n0   = (blockIdx.x * 8 + wave) * 16;
  const bool act = (n0 < N);

  const float* Bgp = Bg + (size_t)(2 * hi) * N + n0 + loN;
  const float* Bup = Bu + (size_t)(2 * hi) * N + n0 + loN;
  const size_t bstep = (size_t)4 * N;

  v8f g0 = {}, g1 = {}, u0 = {}, u1 = {};
  for (int k0 = 0; k0 < K; k0 += CHUNK) {
    stage_async(A, As, K, k0);
    __syncthreads();
    if (act) {
      const float* Ap = &As[loN * LDST];
#pragma unroll 4
      for (int kk = 0; kk < CHUNK; kk += 4) {
        int ka = kk + 2 * hi;
        v2f a0 = *(const v2f*)(Ap + ka);
        v2f a1 = *(const v2f*)(Ap + 16 * LDST + ka);
        v2f bg, bu;
        bg.x = Bgp[0];
        bg.y = Bgp[N];
        bu.x = Bup[0];
        bu.y = Bup[N];
        Bgp += bstep;
        Bup += bstep;
        g0 = __builtin_amdgcn_wmma_f32_16x16x4_f32(false, a0, false, bg, (short)0, g0, false, false);
        g1 = __builtin_amdgcn_wmma_f32_16x16x4_f32(false, a1, false, bg, (short)0, g1, false, false);
        u0 = __builtin_amdgcn_wmma_f32_16x16x4_f32(false, a0, false, bu, (short)0, u0, false, false);
        u1 = __builtin_amdgcn_wmma_f32_16x16x4_f32(false, a1, false, bu, (short)0, u1, false, false);
      }
    }
    __syncthreads();
  }
  if (act) {
#pragma unroll
    for (int r = 0; r < 8; ++r) {
      int m0 = r + 8 * hi;
      int m1 = m0 + 16;
      float ga = g0[r], gb = g1[r];
      float sa = ga / (1.f + __expf(-ga));
      float sb = gb / (1.f + __expf(-gb));
      T1[(size_t)m0 * N + n0 + loN] = sa * u0[r];
      T1[(size_t)m1 * N + n0 + loN] = sb * u1[r];
    }
  }
}

// ---------------------------------------------------------------------------
// RoPE on q (18x1024) and k (18x128). One thread per (row, head, d<32) pair.
// cos_q/sin_q already carry the DH^-0.25 scale.
// ---------------------------------------------------------------------------
__global__ void rope_kernel(float* __restrict__ q, float* __restrict__ k,
                            const float* __restrict__ cos_q,
                            const float* __restrict__ sin_q) {
  int idx = blockIdx.x * blockDim.x + threadIdx.x;   // ROWS * 512
  if (idx >= ROWS * 512) return;
  int m     = idx >> 9;
  int cpair = idx & 511;
  int head  = cpair >> 5;
  int d     = cpair & 31;
  int tok   = m % QLEN;
  float c1 = cos_q[tok * DHEAD + d],      s1 = sin_q[tok * DHEAD + d];
  float c2 = cos_q[tok * DHEAD + d + 32], s2 = sin_q[tok * DHEAD + d + 32];
  float* qp = q + (size_t)m * HDIM + head * DHEAD;
  float x1 = qp[d], x2 = qp[d + 32];
  qp[d]      = x1 * c1 - x2 * s1;
  qp[d + 32] = x2 * c2 + x1 * s2;
  if (head < KVH) {
    float* kp = k + (size_t)m * (KVH * DHEAD) + head * DHEAD;
    x1 = kp[d]; x2 = kp[d + 32];
    kp[d]      = x1 * c1 - x2 * s1;
    kp[d + 32] = x2 * c2 + x1 * s2;
  }
}

// ---------------------------------------------------------------------------
// Attention: one thread per (batch, head, query) -> 288 threads. 9x9 softmax.
// GQA: kv head = head/8.
// ---------------------------------------------------------------------------
__global__ void attn_kernel(const float* __restrict__ q,
                            const float* __restrict__ k,
                            const float* __restrict__ v,
                            float* __restrict__ ao) {
  int idx = blockIdx.x * blockDim.x + threadIdx.x;
  if (idx >= 2 * NHEAD * QLEN) return;
  int i  = idx % QLEN;
  int hd = (idx / QLEN) % NHEAD;
  int b  = idx / (QLEN * NHEAD);
  int kvh = hd >> 3;
  const float* qp = q + (size_t)(b * QLEN + i) * HDIM + hd * DHEAD;
  float s[QLEN];
  float mx = -1e30f;
#pragma unroll
  for (int j = 0; j < QLEN; ++j) {
    const float* kp = k + (size_t)(b * QLEN + j) * (KVH * DHEAD) + kvh * DHEAD;
    float acc = 0.f;
    for (int d = 0; d < DHEAD; ++d) acc += qp[d] * kp[d];
    s[j] = acc;
    mx = fmaxf(mx, acc);
  }
  float sum = 0.f;
#pragma unroll
  for (int j = 0; j < QLEN; ++j) { s[j] = __expf(s[j] - mx); sum += s[j]; }
  float inv = 1.f / sum;
  float* op = ao + (size_t)(b * QLEN + i) * HDIM + hd * DHEAD;
  for (int d = 0; d < DHEAD; ++d) {
    float acc = 0.f;
#pragma unroll
    for (int j = 0; j < QLEN; ++j)
      acc += s[j] * v[(size_t)(b * QLEN + j) * (KVH * DHEAD) + kvh * DHEAD + d];
    op[d] = acc * inv;
  }
}

// ---------------------------------------------------------------------------
// CFG epilogue: st = <pos,neg>/(<neg,neg>+eps); out = random - dt*dphi.
// d_out[0] = step+1 (int bits), d_out[1..256] = next_random.
// ---------------------------------------------------------------------------
__global__ void cfg_kernel(const float* __restrict__ outf,
                           const float* __restrict__ random,
                           const float* __restrict__ dt_all,
                           const float* __restrict__ cfg,
                           const float* __restrict__ cfgm,
                           const int*   __restrict__ step,
                           float* __restrict__ out) {
  __shared__ float redp[256], redn[256];
  int t = threadIdx.x;
  float p = outf[t];          // rows 0..3 flat
  float n = outf[256 + t];    // rows 4..7 flat
  redp[t] = p * n;
  redn[t] = n * n;
  __syncthreads();
  for (int s2 = 128; s2 > 0; s2 >>= 1) {
    if (t < s2) { redp[t] += redp[t + s2]; redn[t] += redn[t + s2]; }
    __syncthreads();
  }
  int s = step[0];
  float st = redp[0] / (redn[0] + 1e-7f);
  float dphi = cfgm[0] * n * st + cfg[0] * p;
  out[1 + t] = random[t] - dt_all[s] * dphi;
  if (t == 0) ((int*)out)[0] = s + 1;
}

// ---------------------------------------------------------------------------
extern "C" void kernel_launch(void* const* d_in, const int* in_sizes, int n_in,
                              void* d_out, int out_size, void* d_ws, size_t ws_size,
                              hipStream_t stream) {
  const float* random     = (const float*)d_in[0];
  const float* dit_hidden = (const float*)d_in[1];
  const float* feat_cond  = (const float*)d_in[2];
  const float* cfg_value  = (const float*)d_in[3];
  const float* cfg_minus  = (const float*)d_in[4];
  const float* t_all      = (const float*)d_in[5];
  const float* dt_all     = (const float*)d_in[6];
  const float* cos_q      = (const float*)d_in[7];
  const float* sin_q      = (const float*)d_in[8];
  const float* W_in       = (const float*)d_in[9];
  const float* b_in       = (const float*)d_in[10];
  const float* ln1_w      = (const float*)d_in[11];
  const float* Wq         = (const float*)d_in[12];
  const float* Wk         = (const float*)d_in[13];
  const float* Wv         = (const float*)d_in[14];
  const float* Wo         = (const float*)d_in[15];
  const float* ln2_w      = (const float*)d_in[16];
  const float* Wg         = (const float*)d_in[17];
  const float* Wu         = (const float*)d_in[18];
  const float* Wd         = (const float*)d_in[19];
  const float* norm_w     = (const float*)d_in[20];
  const float* W_out      = (const float*)d_in[21];
  const float* b_out      = (const float*)d_in[22];
  const int*   step       = (const int*)d_in[23];

  float* ws   = (float*)d_ws;
  float* h    = ws;                      // 32*1024
  float* hn   = h    + MPAD * HDIM;      // 32*1024
  float* q    = hn   + MPAD * HDIM;      // 32*1024
  float* kb   = q    + MPAD * HDIM;      // 32*128
  float* vb   = kb   + MPAD * KVH * DHEAD;
  float* ao   = vb   + MPAD * KVH * DHEAD;   // 32*1024
  float* t1   = ao   + MPAD * HDIM;      // 32*4096
  float* hf   = t1   + MPAD * IDIM;      // 32*1024
  float* outf = hf   + MPAD * HDIM;      // 32*64

  prep_kernel<<<HDIM / 256, 256, 0, stream>>>(random, dit_hidden, feat_cond,
                                              t_all, W_in, b_in, step, h, ao);

  for (int l = 0; l < 8; ++l) {
    const float* wq = Wq + (size_t)l * HDIM * (NHEAD * DHEAD);
    const float* wk = Wk + (size_t)l * HDIM * (KVH * DHEAD);
    const float* wv = Wv + (size_t)l * HDIM * (KVH * DHEAD);
    const float* wo = Wo + (size_t)l * (NHEAD * DHEAD) * HDIM;
    const float* wg = Wg + (size_t)l * HDIM * IDIM;
    const float* wu = Wu + (size_t)l * HDIM * IDIM;
    const float* wd = Wd + (size_t)l * IDIM * HDIM;

    rms_kernel<<<MPAD, 256, 0, stream>>>(h, ln1_w + l * HDIM, hn);
    gemm_kernel<<<8, 256, 0, stream>>>(hn, wq, q,  nullptr, HDIM, NHEAD * DHEAD, 0);
    gemm_kernel<<<1, 256, 0, stream>>>(hn, wk, kb, nullptr, HDIM, KVH * DHEAD, 0);
    gemm_kernel<<<1, 256, 0, stream>>>(hn, wv, vb, nullptr, HDIM, KVH * DHEAD, 0);
    rope_kernel<<<(ROWS * 512 + 255) / 256, 256, 0, stream>>>(q, kb, cos_q, sin_q);
    attn_kernel<<<2, 256, 0, stream>>>(q, kb, vb, ao);
    gemm_kernel<<<8, 256, 0, stream>>>(ao, wo, h, nullptr, HDIM, HDIM, 1);   // h += ao@Wo
    rms_kernel<<<MPAD, 256, 0, stream>>>(h, ln2_w + l * HDIM, hn);
    gemm_gate_up_kernel<<<IDIM / 128, 256, 0, stream>>>(hn, wg, wu, t1, HDIM, IDIM);
    gemm_kernel<<<8, 256, 0, stream>>>(t1, wd, h, nullptr, IDIM, HDIM, 1);   // h += t1@Wd
  }

  rms_final_kernel<<<MPAD, 256, 0, stream>>>(h, norm_w, hf);
  gemm_kernel<<<1, 256, 0, stream>>>(hf, W_out, outf, b_out, HDIM, 64, 2);
  cfg_kernel<<<1, 256, 0, stream>>>(outf, random, dt_all, cfg_value, cfg_minus,
                                    step, (float*)d_out);
}